// MyParticleNetwork_4647154614499
// MI455X (gfx1250) — compile-verified
//
#include <hip/hip_runtime.h>

typedef __attribute__((ext_vector_type(2))) float v2f;
typedef __attribute__((ext_vector_type(8))) float v8f;

#define DT_F 0.02f
#define INV_DT 50.0f
#define INV_RADIUS (1.0f/0.1125f)
#define FOPI 1.2732395447351628f
#define CCONV_BLOCK 128
#define NKERN 64
#define LDS_PAD 4

__device__ __forceinline__ float sgnf(float v) {
    return (v > 0.f) ? 1.f : ((v < 0.f) ? -1.f : 0.f);
}

__device__ __forceinline__ void ball_to_cube(float x, float y, float z,
                                             float& xq, float& yq, float& zc) {
    float sq    = x * x + y * y + z * z;
    float norm  = sqrtf(sq + 1e-24f);
    float xy_sq = x * x + y * y;
    bool  zero  = sq < 1e-12f;
    bool  cone  = 1.25f * z * z > xy_sq;
    float s_cone = sqrtf(3.f * norm / (norm + fabsf(z) + 1e-12f));
    float s_side = norm * rsqrtf(xy_sq + 1e-24f);
    float xc = zero ? 0.f : (cone ? x * s_cone : x * s_side);
    float yc = zero ? 0.f : (cone ? y * s_cone : y * s_side);
    zc       = zero ? 0.f : (cone ? sgnf(z) * norm : 1.5f * z);
    float nxy_sq = xc * xc + yc * yc;
    float nxy    = sqrtf(nxy_sq + 1e-24f);
    bool  zero_xy = nxy_sq < 1e-12f;
    bool  xbig    = fabsf(xc) > fabsf(yc);
    float dx_ = (fabsf(xc) > 1e-12f) ? xc : 1.f;
    float dy_ = (fabsf(yc) > 1e-12f) ? yc : 1.f;
    float tx = sgnf(xc) * nxy;
    float ty = sgnf(yc) * nxy;
    xq = zero_xy ? 0.f : (xbig ? tx : ty * FOPI * atanf(xc / dy_));
    yq = zero_xy ? 0.f : (xbig ? tx * FOPI * atanf(yc / dx_) : ty);
}

// ---------------------------------------------------------------------------
// Fused ContinuousConv per 16-dst tile. LDS layout: T[row][ci][bin] with
// padded row stride (CH*64 + LDS_PAD). K flattening: kk = ci*64 + bin.
// W row for kk: (bin*Cin + c0 + ci). Cin must be a multiple of CH.
// ---------------------------------------------------------------------------
template <int CH>
__global__ __launch_bounds__(CCONV_BLOCK)
void cconv_kernel(const float* __restrict__ sf, int Cin,
                  const float* __restrict__ sp, const float* __restrict__ dp,
                  const int* __restrict__ esrc, const int* __restrict__ edst, int nE,
                  const float* __restrict__ W, const float* __restrict__ bias,
                  float* __restrict__ out, int ldo, int col_off, int Cout, int nDst) {
    constexpr int STRIDE = CH * NKERN + LDS_PAD;
    extern __shared__ float sT[];   // 16 * STRIDE floats

    const int d0r  = blockIdx.x * 16;
    const int tid  = threadIdx.x;
    const int lane = tid & 31;
    const int wv   = tid >> 5;

    // edge range for dst rows [d0r, d0r+16) -- edst is sorted ascending
    int lo, hi;
    {
        int a = 0, b = nE;
        while (a < b) { int m = (a + b) >> 1; if (edst[m] < d0r) a = m + 1; else b = m; }
        lo = a;
        b = nE;
        while (a < b) { int m = (a + b) >> 1; if (edst[m] < d0r + 16) a = m + 1; else b = m; }
        hi = a;
    }

    v8f acc = {0.f, 0.f, 0.f, 0.f, 0.f, 0.f, 0.f, 0.f};
    const int  n0     = wv * 16;
    const bool doGemm = (n0 < Cout);                    // wave-uniform
    // clamped output column: OOB columns compute garbage that is never stored
    const int  colc   = min(n0 + (lane & 15), Cout - 1);
    const int  rowA   = lane & 15;
    const int  khalf  = (lane >> 4) << 1;               // 0 or 2

    const int nch = Cin / CH;
    for (int chn = 0; chn < nch; ++chn) {
        const int c0 = chn * CH;
        for (int i = tid; i < 16 * STRIDE; i += CCONV_BLOCK) sT[i] = 0.f;
        __syncthreads();

        // ---- scatter edges into LDS T ----
        for (int e = lo + tid; e < hi; e += CCONV_BLOCK) {
            const int s = esrc[e];
            const int d = edst[e];
            const float rx = (sp[s * 3 + 0] - dp[d * 3 + 0]) * INV_RADIUS;
            const float ry = (sp[s * 3 + 1] - dp[d * 3 + 1]) * INV_RADIUS;
            const float rz = (sp[s * 3 + 2] - dp[d * 3 + 2]) * INV_RADIUS;
            const float r2 = rx * rx + ry * ry + rz * rz;
            float om  = 1.f - r2;
            float win = fminf(fmaxf(om * om * om, 0.f), 1.f);

            float bx, by, bz;
            ball_to_cube(rx, ry, rz, bx, by, bz);
            float gx = (bx + 1.f) * 1.5f;
            float gy = (by + 1.f) * 1.5f;
            float gz = (bz + 1.f) * 1.5f;
            float fx = fminf(fmaxf(floorf(gx), 0.f), 2.f);
            float fy = fminf(fmaxf(floorf(gy), 0.f), 2.f);
            float fz = fminf(fmaxf(floorf(gz), 0.f), 2.f);
            const int ix = (int)fx, iy = (int)fy, iz = (int)fz;
            const float frx = gx - fx, fry = gy - fy, frz = gz - fz;

            int   bins[8];
            float wts[8];
            {
                int q = 0;
                for (int dz = 0; dz < 2; ++dz) {
                    const float wz = dz ? frz : 1.f - frz;
                    for (int dy = 0; dy < 2; ++dy) {
                        const float wy = dy ? fry : 1.f - fry;
                        for (int dx = 0; dx < 2; ++dx) {
                            const float wx = dx ? frx : 1.f - frx;
                            bins[q] = ((iz + dz) * 4 + (iy + dy)) * 4 + (ix + dx);
                            wts[q]  = wz * wy * wx;
                            ++q;
                        }
                    }
                }
            }
            float* Trow0 = &sT[(d - d0r) * STRIDE];
            const float* fp = sf + (size_t)s * Cin + c0;
            if constexpr ((CH & 3) == 0) {
                const float4* fp4 = (const float4*)fp;
#pragma unroll
                for (int c4 = 0; c4 < CH / 4; ++c4) {
                    const float4 f4 = fp4[c4];
                    const float fv[4] = {f4.x * win, f4.y * win, f4.z * win, f4.w * win};
#pragma unroll
                    for (int j = 0; j < 4; ++j) {
                        float* Tc = Trow0 + (c4 * 4 + j) * NKERN;
#pragma unroll
                        for (int t = 0; t < 8; ++t)
                            atomicAdd(&Tc[bins[t]], wts[t] * fv[j]);
                    }
                }
            } else {
#pragma unroll
                for (int c = 0; c < CH; ++c) {
                    const float fv = fp[c] * win;
                    float* Tc = Trow0 + c * NKERN;
#pragma unroll
                    for (int t = 0; t < 8; ++t)
                        atomicAdd(&Tc[bins[t]], wts[t] * fv);
                }
            }
        }
        __syncthreads();

        // ---- WMMA GEMM over this chunk: K = 64*CH, kk = ci*64 + bin ----
        if (doGemm) {
            constexpr int Ktot = NKERN * CH;
            const float* Trow = &sT[rowA * STRIDE];
            const float* Wcol = W + (size_t)(c0)*Cout + colc;

            auto loadA = [&](int kk) -> v2f {
                return *(const v2f*)&Trow[kk];
            };
            auto loadB = [&](int kk) -> v2f {
                const int bin = kk & 63;
                const int ci  = kk >> 6;
                const size_t idx = (size_t)(bin * Cin + ci) * Cout;
                v2f b;
                b[0] = Wcol[idx];
                b[1] = Wcol[idx + (size_t)Cin * Cout];   // bin+1, same ci
                return b;
            };

            v2f a = loadA(khalf);
            v2f b = loadB(khalf);
            int k0 = 0;
#pragma unroll 2
            for (; k0 + 4 < Ktot; k0 += 4) {
                const int kn = k0 + 4 + khalf;
                v2f an = loadA(kn);
                v2f bn = loadB(kn);
                acc = __builtin_amdgcn_wmma_f32_16x16x4_f32(
                    false, a, false, b, (short)0, acc, false, false);
                a = an; b = bn;
            }
            acc = __builtin_amdgcn_wmma_f32_16x16x4_f32(
                false, a, false, b, (short)0, acc, false, false);
        }
        __syncthreads();
    }

    // ---- store C tile + bias ----
    if (doGemm) {
        const int N    = lane & 15;
        const int mhi  = (lane >> 4) * 8;
        const int coln = n0 + N;
        if (coln < Cout) {
            const float bv = bias[coln];
#pragma unroll
            for (int r = 0; r < 8; ++r) {
                const int grow = d0r + r + mhi;
                if (grow < nDst)
                    out[(size_t)grow * ldo + col_off + coln] = acc[r] + bv;
            }
        }
    }
}

// ---------------------------------------------------------------------------
__global__ void integrate_kernel(const float* __restrict__ pos,
                                 const float* __restrict__ vel,
                                 float* __restrict__ pos2,
                                 float* __restrict__ feats, int n) {
    int i = blockIdx.x * blockDim.x + threadIdx.x;
    if (i >= n) return;
    float vx = vel[i * 3 + 0], vy = vel[i * 3 + 1], vz = vel[i * 3 + 2];
    float v2x = vx, v2y = vy - 9.81f * DT_F, v2z = vz;
    pos2[i * 3 + 0] = pos[i * 3 + 0] + DT_F * (v2x + vx) * 0.5f;
    pos2[i * 3 + 1] = pos[i * 3 + 1] + DT_F * (v2y + vy) * 0.5f;
    pos2[i * 3 + 2] = pos[i * 3 + 2] + DT_F * (v2z + vz) * 0.5f;
    feats[i * 4 + 0] = 1.f;
    feats[i * 4 + 1] = v2x;
    feats[i * 4 + 2] = v2y;
    feats[i * 4 + 3] = v2z;
}

// y[row*ldo+coff+col] = x[row,:]@Wd[:,col] + bd[col] (+add0) (+add1)
__global__ void dense_kernel(const float* __restrict__ x, int Cin,
                             const float* __restrict__ Wd, const float* __restrict__ bd,
                             const float* __restrict__ add0, const float* __restrict__ add1,
                             float* __restrict__ y, int n, int Cout, int ldo, int coff) {
    int idx = blockIdx.x * blockDim.x + threadIdx.x;
    if (idx >= n * Cout) return;
    int row = idx / Cout;
    int col = idx - row * Cout;
    float s = bd[col];
    const float* xr = x + (size_t)row * Cin;
    for (int c = 0; c < Cin; ++c) s += xr[c] * Wd[c * Cout + col];
    if (add0) s += add0[(size_t)row * Cout + col];
    if (add1) s += add1[(size_t)row * Cout + col];
    y[(size_t)row * ldo + coff + col] = s;
}

__global__ void relu_kernel(const float* __restrict__ in, float* __restrict__ out, int n) {
    int i = blockIdx.x * blockDim.x + threadIdx.x;
    if (i < n) out[i] = fmaxf(in[i], 0.f);
}

__global__ void finalize_kernel(const float* __restrict__ pos,
                                const float* __restrict__ pos2,
                                const float* __restrict__ h3,
                                float* __restrict__ out, int n) {
    int i = blockIdx.x * blockDim.x + threadIdx.x;
    if (i >= n) return;
#pragma unroll
    for (int ax = 0; ax < 3; ++ax) {
        float pn = pos2[i * 3 + ax] + h3[i * 3 + ax] * (1.f / 128.f);
        out[i * 6 + ax]     = pn;
        out[i * 6 + 3 + ax] = (pn - pos[i * 3 + ax]) * INV_DT;
    }
}

// ---------------------------------------------------------------------------
extern "C" void kernel_launch(void* const* d_in, const int* in_sizes, int n_in,
                              void* d_out, int out_size, void* d_ws, size_t ws_size,
                              hipStream_t stream) {
    const float* pos       = (const float*)d_in[0];
    const float* vel       = (const float*)d_in[1];
    const float* box       = (const float*)d_in[2];
    const float* box_feats = (const float*)d_in[3];
    const int*   ff_src    = (const int*)d_in[4];
    const int*   ff_dst    = (const int*)d_in[5];
    const int*   bf_src    = (const int*)d_in[6];
    const int*   bf_dst    = (const int*)d_in[7];
    const float* W0f = (const float*)d_in[8];
    const float* b0f = (const float*)d_in[9];
    const float* W0o = (const float*)d_in[10];
    const float* b0o = (const float*)d_in[11];
    const float* Wd0 = (const float*)d_in[12];
    const float* bd0 = (const float*)d_in[13];
    const float* Wc1 = (const float*)d_in[14];
    const float* bc1 = (const float*)d_in[15];
    const float* Wd1 = (const float*)d_in[16];
    const float* bd1 = (const float*)d_in[17];
    const float* Wc2 = (const float*)d_in[18];
    const float* bc2 = (const float*)d_in[19];
    const float* Wd2 = (const float*)d_in[20];
    const float* bd2 = (const float*)d_in[21];
    const float* Wc3 = (const float*)d_in[22];
    const float* bc3 = (const float*)d_in[23];
    const float* Wd3 = (const float*)d_in[24];
    const float* bd3 = (const float*)d_in[25];

    const int n     = in_sizes[0] / 3;
    const int nE_ff = in_sizes[4];
    const int nE_bf = in_sizes[6];

    // workspace layout (fp32)
    float* w     = (float*)d_ws;
    float* pos2  = w;  w += (size_t)n * 3;
    float* feats = w;  w += (size_t)n * 4;
    float* h     = w;  w += (size_t)n * 96;   // layer0 concat, reused for h3
    float* x     = w;  w += (size_t)n * 96;   // relu buffer (reused each layer)
    float* cc    = w;  w += (size_t)n * 64;   // cconv output scratch
    float* h1    = w;  w += (size_t)n * 64;
    float* h2    = w;  w += (size_t)n * 64;
    (void)ws_size; (void)n_in; (void)out_size;

    const size_t lds32 = (size_t)16 * (32 * NKERN + LDS_PAD) * sizeof(float);
    const size_t lds4  = (size_t)16 * (4  * NKERN + LDS_PAD) * sizeof(float);
    const size_t lds3  = (size_t)16 * (3  * NKERN + LDS_PAD) * sizeof(float);
    (void)hipFuncSetAttribute(reinterpret_cast<const void*>(cconv_kernel<32>),
                              hipFuncAttributeMaxDynamicSharedMemorySize, (int)lds32);
    (void)hipFuncSetAttribute(reinterpret_cast<const void*>(cconv_kernel<4>),
                              hipFuncAttributeMaxDynamicSharedMemorySize, (int)lds4);
    (void)hipFuncSetAttribute(reinterpret_cast<const void*>(cconv_kernel<3>),
                              hipFuncAttributeMaxDynamicSharedMemorySize, (int)lds3);

    const int tiles = (n + 15) / 16;

    integrate_kernel<<<(n + 255) / 256, 256, 0, stream>>>(pos, vel, pos2, feats, n);

    // layer 0: h = [cconv(box_feats) | cconv(feats) | feats@Wd0]
    cconv_kernel<3><<<tiles, CCONV_BLOCK, lds3, stream>>>(
        box_feats, 3, box, pos2, bf_src, bf_dst, nE_bf, W0o, b0o, h, 96, 0, 32, n);
    cconv_kernel<4><<<tiles, CCONV_BLOCK, lds4, stream>>>(
        feats, 4, pos2, pos2, ff_src, ff_dst, nE_ff, W0f, b0f, h, 96, 32, 32, n);
    dense_kernel<<<(n * 32 + 255) / 256, 256, 0, stream>>>(
        feats, 4, Wd0, bd0, nullptr, nullptr, h, n, 32, 96, 64);
    relu_kernel<<<(n * 96 + 255) / 256, 256, 0, stream>>>(h, x, n * 96);

    // layer 1: h1 = cconv(x) + x@Wd1 + bd1
    cconv_kernel<32><<<tiles, CCONV_BLOCK, lds32, stream>>>(
        x, 96, pos2, pos2, ff_src, ff_dst, nE_ff, Wc1, bc1, cc, 64, 0, 64, n);
    dense_kernel<<<(n * 64 + 255) / 256, 256, 0, stream>>>(
        x, 96, Wd1, bd1, cc, nullptr, h1, n, 64, 64, 0);
    relu_kernel<<<(n * 64 + 255) / 256, 256, 0, stream>>>(h1, x, n * 64);

    // layer 2: h2 = cconv(x) + x@Wd2 + bd2 + h1
    cconv_kernel<32><<<tiles, CCONV_BLOCK, lds32, stream>>>(
        x, 64, pos2, pos2, ff_src, ff_dst, nE_ff, Wc2, bc2, cc, 64, 0, 64, n);
    dense_kernel<<<(n * 64 + 255) / 256, 256, 0, stream>>>(
        x, 64, Wd2, bd2, cc, h1, h2, n, 64, 64, 0);
    relu_kernel<<<(n * 64 + 255) / 256, 256, 0, stream>>>(h2, x, n * 64);

    // layer 3: h3 = cconv(x) + x@Wd3 + bd3   (Cout = 3), stored in h
    cconv_kernel<32><<<tiles, CCONV_BLOCK, lds32, stream>>>(
        x, 64, pos2, pos2, ff_src, ff_dst, nE_ff, Wc3, bc3, cc, 3, 0, 3, n);
    dense_kernel<<<(n * 3 + 255) / 256, 256, 0, stream>>>(
        x, 64, Wd3, bd3, cc, nullptr, h, n, 3, 3, 0);

    finalize_kernel<<<(n + 255) / 256, 256, 0, stream>>>(pos, pos2, h, (float*)d_out, n);
}